// RocketLeagueGAT_17867063951808
// MI455X (gfx1250) — compile-verified
//
#include <hip/hip_runtime.h>
#include <math.h>

typedef float v2f __attribute__((ext_vector_type(2)));
typedef float v8f __attribute__((ext_vector_type(8)));

// ---------------- problem constants ----------------
constexpr int NGr  = 8192;          // graphs
constexpr int Pn   = 7;             // nodes per graph
constexpr int NN   = NGr * Pn;      // 57344 nodes
constexpr int FPi  = 16;            // input features
constexpr int GFd  = 10;            // global features
constexpr int HID  = 64;
constexpr int H1n  = 4;
constexpr int C1   = H1n * HID;     // 256
constexpr int EPG  = Pn * (Pn - 1); // 42 edges per graph
constexpr int EDd  = 4;

// BN partial-reduction geometry (deterministic two-pass)
constexpr int PB   = 512;           // partial blocks
constexpr int RPB  = NN / PB;       // 112 rows per block (exact)

// ---------------- workspace layout (floats) ----------------
constexpr size_t OFF_H1    = 0;                         // N*256 (later aliased)
constexpr size_t OFF_HAGG1 = OFF_H1 + (size_t)NN * C1;  // N*256
constexpr size_t OFF_ALS1  = OFF_HAGG1 + (size_t)NN * C1;
constexpr size_t OFF_ALD1  = OFF_ALS1 + (size_t)NN * H1n;
constexpr size_t OFF_V1    = OFF_ALD1 + (size_t)NN * H1n;   // 16
constexpr size_t OFF_V2    = OFF_V1 + 16;                   // 4
constexpr size_t OFF_SUM1  = OFF_V2 + 4;                    // PB*256
constexpr size_t OFF_SQ1   = OFF_SUM1 + (size_t)PB * C1;
constexpr size_t OFF_SC1   = OFF_SQ1 + (size_t)PB * C1;     // 256 (scale)
constexpr size_t OFF_SH1   = OFF_SC1 + C1;                  // 256 (shift)
constexpr size_t OFF_SUM2  = OFF_SH1 + C1;                  // PB*64
constexpr size_t OFF_SQ2   = OFF_SUM2 + (size_t)PB * HID;
constexpr size_t OFF_SC2   = OFF_SQ2 + (size_t)PB * HID;    // 64
constexpr size_t OFF_SH2   = OFF_SC2 + HID;                 // 64
// layer-2 tensors alias the dead h1 region:
constexpr size_t OFF_H2    = OFF_H1;                        // N*64
constexpr size_t OFF_HAGG2 = OFF_H1 + (size_t)NN * HID;     // N*64
constexpr size_t OFF_ALS2  = OFF_H1 + (size_t)2 * NN * HID; // N
constexpr size_t OFF_ALD2  = OFF_ALS2 + NN;                 // N

// local edge e -> (src, dst) within a 7-node graph (meshgrid order, s != d)
__device__ __forceinline__ void edge_sd(int e, int& s, int& d) {
    s = e / (Pn - 1);
    int r = e - s * (Pn - 1);
    d = r + (r >= s ? 1 : 0);
}
__device__ __forceinline__ int edge_id(int s, int d) {  // inverse, s != d
    return s * (Pn - 1) + d - (d > s ? 1 : 0);
}

// ---------------- tiny precompute: v1[h][d], v2[d] ----------------
__global__ void k_precompute_v(const float* __restrict__ We1, const float* __restrict__ ae1,
                               const float* __restrict__ We2, const float* __restrict__ ae2,
                               float* __restrict__ v1, float* __restrict__ v2) {
    int t = threadIdx.x;
    if (t < 16) {
        int h = t >> 2, d = t & 3;
        float s = 0.f;
        for (int c = 0; c < HID; ++c) s += We1[d * C1 + h * HID + c] * ae1[h * HID + c];
        v1[t] = s;                      // v1[h*4+d]
    } else if (t < 20) {
        int d = t - 16;
        float s = 0.f;
        for (int c = 0; c < HID; ++c) s += We2[d * HID + c] * ae2[c];
        v2[d] = s;
    }
}

// ---------------- GEMM1: h1 = x @ W1  ([N,16]x[16,256]) via WMMA f32 ----------------
// Each wave computes a 16x64 output slab (4 N-tiles) reusing one A fragment.
__global__ void k_gemm1_wmma(const float* __restrict__ x, const float* __restrict__ W1,
                             float* __restrict__ h1) {
    const int lane   = threadIdx.x & 31;
    const int wave   = threadIdx.x >> 5;
    const int wgl    = blockIdx.x * 8 + wave;   // (N/16)*(256/64) waves total
    const int tm     = wgl >> 2;
    const int tq     = wgl & 3;                 // which 64-col slab
    const int laneLo = lane & 15;
    const int laneHi = lane >> 4;
    const int row    = tm * 16 + laneLo;
    const int colB   = tq * 64 + laneLo;
    v8f acc[4] = {};
#pragma unroll
    for (int kk = 0; kk < 4; ++kk) {
        const int k = kk * 4 + laneHi * 2;
        const v2f a = *(const v2f*)(x + (size_t)row * FPi + k);   // 8B-aligned pair
#pragma unroll
        for (int j = 0; j < 4; ++j) {
            v2f b;
            b.x = W1[k * C1 + colB + j * 16];
            b.y = W1[(k + 1) * C1 + colB + j * 16];
            acc[j] = __builtin_amdgcn_wmma_f32_16x16x4_f32(false, a, false, b, (short)0,
                                                           acc[j], false, false);
        }
    }
#pragma unroll
    for (int j = 0; j < 4; ++j)
#pragma unroll
        for (int r = 0; r < 8; ++r)
            h1[(size_t)(tm * 16 + r + laneHi * 8) * C1 + colB + j * 16] = acc[j][r];
}

// ---------------- per-node attention logits, layer 1 ----------------
__global__ void k_nodelogits1(const float* __restrict__ h1, const float* __restrict__ as1,
                              const float* __restrict__ ad1, float* __restrict__ als,
                              float* __restrict__ ald) {
    int t = blockIdx.x * blockDim.x + threadIdx.x;   // N*H1 threads
    int n = t >> 2, h = t & 3;
    const float* base = h1 + (size_t)n * C1 + h * HID;
    float s = 0.f, d = 0.f;
    for (int c = 0; c < HID; ++c) {
        float v = base[c];
        s += v * as1[h * HID + c];
        d += v * ad1[h * HID + c];
    }
    als[t] = s;
    ald[t] = d;
}

// ---------------- layer-1 GAT: softmax over 6 in-edges + aggregate (+b1) ----------------
__global__ void k_gat1(const float* __restrict__ h1, const float* __restrict__ als,
                       const float* __restrict__ ald, const float* __restrict__ eattr,
                       const float* __restrict__ v1, const float* __restrict__ b1,
                       float* __restrict__ hagg1) {
    __shared__ float sh_h[Pn * C1];
    __shared__ float sh_as[Pn * H1n], sh_ad[Pn * H1n];
    __shared__ float sh_l[EPG * H1n];
    const int g = blockIdx.x;
    const int t = threadIdx.x;

    for (int i = 0; i < Pn; ++i)
        sh_h[i * C1 + t] = h1[(size_t)(g * Pn + i) * C1 + t];
    if (t < Pn * H1n) {
        sh_as[t] = als[g * Pn * H1n + t];
        sh_ad[t] = ald[g * Pn * H1n + t];
    }
    __syncthreads();

    if (t < EPG * H1n) {
        int e = t >> 2, h = t & 3;
        int s, d; edge_sd(e, s, d);
        const float* ea = eattr + (size_t)(g * EPG + e) * EDd;
        float ale = ea[0] * v1[h * 4 + 0] + ea[1] * v1[h * 4 + 1] +
                    ea[2] * v1[h * 4 + 2] + ea[3] * v1[h * 4 + 3];
        float l = sh_as[s * H1n + h] + sh_ad[d * H1n + h] + ale;
        sh_l[t] = (l > 0.f) ? l : 0.2f * l;   // leaky_relu 0.2
    }
    __syncthreads();

    if (t < Pn * H1n) {           // segment softmax per (dst,head)
        int d = t >> 2, h = t & 3;
        float mx = -1e30f;
        for (int s = 0; s < Pn; ++s)
            if (s != d) mx = fmaxf(mx, sh_l[edge_id(s, d) * H1n + h]);
        float z = 0.f;
        float p[Pn];
        for (int s = 0; s < Pn; ++s)
            if (s != d) { float v = expf(sh_l[edge_id(s, d) * H1n + h] - mx); p[s] = v; z += v; }
        float inv = 1.f / (z + 1e-16f);
        for (int s = 0; s < Pn; ++s)
            if (s != d) sh_l[edge_id(s, d) * H1n + h] = p[s] * inv;  // alpha in place
    }
    __syncthreads();

    const int h = t >> 6;         // t = h*64 + c  (channel within concat layout)
    const float bias = b1[t];
    for (int d = 0; d < Pn; ++d) {
        float acc = 0.f;
        for (int s = 0; s < Pn; ++s)
            if (s != d) acc = fmaf(sh_l[edge_id(s, d) * H1n + h], sh_h[s * C1 + t], acc);
        hagg1[(size_t)(g * Pn + d) * C1 + t] = acc + bias;
    }
}

// ---------------- BatchNorm partial sums (deterministic, no atomics) ----------------
template <int CH>
__global__ void k_bn_partial(const float* __restrict__ v, float* __restrict__ sum,
                             float* __restrict__ sq) {
    const int c = threadIdx.x;          // CH threads
    const int r0 = blockIdx.x * RPB;
    float s = 0.f, ss = 0.f;
    for (int r = 0; r < RPB; ++r) {
        float x = v[(size_t)(r0 + r) * CH + c];
        s += x;
        ss += x * x;
    }
    sum[blockIdx.x * CH + c] = s;
    sq[blockIdx.x * CH + c] = ss;
}

template <int CH>
__global__ void k_bn_final(const float* __restrict__ sum, const float* __restrict__ sq,
                           const float* __restrict__ gam, const float* __restrict__ bet,
                           float* __restrict__ scale, float* __restrict__ shift) {
    const int c = threadIdx.x;
    float S = 0.f, SS = 0.f;
    for (int b = 0; b < PB; ++b) { S += sum[b * CH + c]; SS += sq[b * CH + c]; }
    const float mean = S / (float)NN;
    const float var = SS / (float)NN - mean * mean;   // biased, as reference
    const float rstd = rsqrtf(var + 1e-5f);
    const float sc = rstd * gam[c];
    scale[c] = sc;
    shift[c] = bet[c] - mean * sc;
}

// ---------------- GEMM2: h2 = relu(bn(hagg1)) @ W2  ([N,256]x[256,64]) via WMMA ----------------
// Each wave computes a full 16x64 output slab; hagg1 is read exactly once,
// and the fused BN+ReLU on the A operand is done exactly once.
__global__ void k_gemm2_wmma(const float* __restrict__ hagg1, const float* __restrict__ W2,
                             const float* __restrict__ scale, const float* __restrict__ shift,
                             float* __restrict__ h2) {
    const int lane   = threadIdx.x & 31;
    const int wave   = threadIdx.x >> 5;
    const int tm     = blockIdx.x * 8 + wave;   // N/16 waves total
    const int laneLo = lane & 15;
    const int laneHi = lane >> 4;
    const int row    = tm * 16 + laneLo;
    v8f acc[4] = {};
    for (int kk = 0; kk < C1 / 4; ++kk) {
        const int k = kk * 4 + laneHi * 2;
        const v2f xv = *(const v2f*)(hagg1 + (size_t)row * C1 + k);  // 8B-aligned pair
        const v2f sv = *(const v2f*)(scale + k);
        const v2f hv = *(const v2f*)(shift + k);
        v2f a;
        a.x = fmaxf(fmaf(xv.x, sv.x, hv.x), 0.f);   // fused BN + ReLU
        a.y = fmaxf(fmaf(xv.y, sv.y, hv.y), 0.f);
#pragma unroll
        for (int j = 0; j < 4; ++j) {
            v2f b;
            b.x = W2[k * HID + j * 16 + laneLo];
            b.y = W2[(k + 1) * HID + j * 16 + laneLo];
            acc[j] = __builtin_amdgcn_wmma_f32_16x16x4_f32(false, a, false, b, (short)0,
                                                           acc[j], false, false);
        }
    }
#pragma unroll
    for (int j = 0; j < 4; ++j)
#pragma unroll
        for (int r = 0; r < 8; ++r)
            h2[(size_t)(tm * 16 + r + laneHi * 8) * HID + j * 16 + laneLo] = acc[j][r];
}

// ---------------- per-node attention logits, layer 2 (H=1) ----------------
__global__ void k_nodelogits2(const float* __restrict__ h2, const float* __restrict__ as2,
                              const float* __restrict__ ad2, float* __restrict__ als,
                              float* __restrict__ ald) {
    int n = blockIdx.x * blockDim.x + threadIdx.x;   // N threads
    const float* base = h2 + (size_t)n * HID;
    float s = 0.f, d = 0.f;
    for (int c = 0; c < HID; ++c) {
        float v = base[c];
        s += v * as2[c];
        d += v * ad2[c];
    }
    als[n] = s;
    ald[n] = d;
}

// ---------------- layer-2 GAT (1 head) ----------------
__global__ void k_gat2(const float* __restrict__ h2, const float* __restrict__ als,
                       const float* __restrict__ ald, const float* __restrict__ eattr,
                       const float* __restrict__ v2, const float* __restrict__ b2,
                       float* __restrict__ hagg2) {
    __shared__ float sh_h[Pn * HID];
    __shared__ float sh_as[Pn], sh_ad[Pn];
    __shared__ float sh_l[EPG];
    const int g = blockIdx.x;
    const int t = threadIdx.x;   // 64 threads

    for (int i = 0; i < Pn; ++i)
        sh_h[i * HID + t] = h2[(size_t)(g * Pn + i) * HID + t];
    if (t < Pn) { sh_as[t] = als[g * Pn + t]; sh_ad[t] = ald[g * Pn + t]; }
    __syncthreads();

    if (t < EPG) {
        int s, d; edge_sd(t, s, d);
        const float* ea = eattr + (size_t)(g * EPG + t) * EDd;
        float ale = ea[0] * v2[0] + ea[1] * v2[1] + ea[2] * v2[2] + ea[3] * v2[3];
        float l = sh_as[s] + sh_ad[d] + ale;
        sh_l[t] = (l > 0.f) ? l : 0.2f * l;
    }
    __syncthreads();

    if (t < Pn) {
        int d = t;
        float mx = -1e30f;
        for (int s = 0; s < Pn; ++s)
            if (s != d) mx = fmaxf(mx, sh_l[edge_id(s, d)]);
        float z = 0.f, p[Pn];
        for (int s = 0; s < Pn; ++s)
            if (s != d) { float v = expf(sh_l[edge_id(s, d)] - mx); p[s] = v; z += v; }
        float inv = 1.f / (z + 1e-16f);
        for (int s = 0; s < Pn; ++s)
            if (s != d) sh_l[edge_id(s, d)] = p[s] * inv;
    }
    __syncthreads();

    const float bias = b2[t];
    for (int d = 0; d < Pn; ++d) {
        float acc = 0.f;
        for (int s = 0; s < Pn; ++s)
            if (s != d) acc = fmaf(sh_l[edge_id(s, d)], sh_h[s * HID + t], acc);
        hagg2[(size_t)(g * Pn + d) * HID + t] = acc + bias;
    }
}

// ---------------- head: BN2+ReLU, mean-pool, concat gf, 2x MLP ----------------
__global__ void k_head(const float* __restrict__ hagg2, const float* __restrict__ scale,
                       const float* __restrict__ shift, const float* __restrict__ gf,
                       const float* __restrict__ Wo1, const float* __restrict__ bo1,
                       const float* __restrict__ Wo2, const float* __restrict__ bo2,
                       const float* __restrict__ Wb1, const float* __restrict__ bb1,
                       const float* __restrict__ Wb2, const float* __restrict__ bb2,
                       float* __restrict__ out) {
    __shared__ float sh_comb[HID + GFd];   // 74
    __shared__ float sh_hid[HID];          // 32 orange + 32 blue
    const int g = blockIdx.x;
    const int t = threadIdx.x;             // 64 threads

    {   // pooled channel t
        const float sc = scale[t], sh = shift[t];
        float acc = 0.f;
        for (int n = 0; n < Pn; ++n) {
            float v = fmaf(hagg2[(size_t)(g * Pn + n) * HID + t], sc, sh);
            acc += fmaxf(v, 0.f);
        }
        sh_comb[t] = acc * (1.f / (float)Pn);
    }
    if (t < GFd) sh_comb[HID + t] = gf[(size_t)g * GFd + t];
    __syncthreads();

    if (t < 32) {              // orange hidden unit t
        float h = bo1[t];
        for (int i = 0; i < HID + GFd; ++i) h = fmaf(sh_comb[i], Wo1[i * 32 + t], h);
        sh_hid[t] = fmaxf(h, 0.f) * Wo2[t];
    } else {                   // blue hidden unit t-32
        int j = t - 32;
        float h = bb1[j];
        for (int i = 0; i < HID + GFd; ++i) h = fmaf(sh_comb[i], Wb1[i * 32 + j], h);
        sh_hid[32 + j] = fmaxf(h, 0.f) * Wb2[j];
    }
    __syncthreads();

    if (t == 0) {
        float s = bo2[0];
        for (int j = 0; j < 32; ++j) s += sh_hid[j];
        out[g] = s;
    } else if (t == 1) {
        float s = bb2[0];
        for (int j = 0; j < 32; ++j) s += sh_hid[32 + j];
        out[NGr + g] = s;
    }
}

// ---------------- launch ----------------
extern "C" void kernel_launch(void* const* d_in, const int* in_sizes, int n_in,
                              void* d_out, int out_size, void* d_ws, size_t ws_size,
                              hipStream_t stream) {
    (void)in_sizes; (void)n_in; (void)out_size; (void)ws_size;
    const float* x    = (const float*)d_in[0];
    const float* ea   = (const float*)d_in[1];
    const float* gf   = (const float*)d_in[2];
    const float* W1   = (const float*)d_in[3];
    const float* as1  = (const float*)d_in[4];
    const float* ad1  = (const float*)d_in[5];
    const float* We1  = (const float*)d_in[6];
    const float* ae1  = (const float*)d_in[7];
    const float* b1   = (const float*)d_in[8];
    const float* g1   = (const float*)d_in[9];
    const float* be1  = (const float*)d_in[10];
    const float* W2   = (const float*)d_in[11];
    const float* as2  = (const float*)d_in[12];
    const float* ad2  = (const float*)d_in[13];
    const float* We2  = (const float*)d_in[14];
    const float* ae2  = (const float*)d_in[15];
    const float* b2   = (const float*)d_in[16];
    const float* g2   = (const float*)d_in[17];
    const float* be2  = (const float*)d_in[18];
    const float* Wo1  = (const float*)d_in[19];
    const float* bo1  = (const float*)d_in[20];
    const float* Wo2  = (const float*)d_in[21];
    const float* bo2  = (const float*)d_in[22];
    const float* Wb1  = (const float*)d_in[23];
    const float* bb1  = (const float*)d_in[24];
    const float* Wb2  = (const float*)d_in[25];
    const float* bb2  = (const float*)d_in[26];
    // d_in[27]=edge_index, d_in[28]=batch : structure is analytic, unused.

    float* ws   = (float*)d_ws;
    float* h1    = ws + OFF_H1;
    float* hagg1 = ws + OFF_HAGG1;
    float* als1  = ws + OFF_ALS1;
    float* ald1  = ws + OFF_ALD1;
    float* v1    = ws + OFF_V1;
    float* v2    = ws + OFF_V2;
    float* sum1  = ws + OFF_SUM1;
    float* sq1   = ws + OFF_SQ1;
    float* sc1   = ws + OFF_SC1;
    float* sh1   = ws + OFF_SH1;
    float* sum2  = ws + OFF_SUM2;
    float* sq2   = ws + OFF_SQ2;
    float* sc2   = ws + OFF_SC2;
    float* sh2   = ws + OFF_SH2;
    float* h2    = ws + OFF_H2;      // aliases dead h1 region
    float* hagg2 = ws + OFF_HAGG2;
    float* als2  = ws + OFF_ALS2;
    float* ald2  = ws + OFF_ALD2;
    float* out   = (float*)d_out;

    k_precompute_v<<<1, 32, 0, stream>>>(We1, ae1, We2, ae2, v1, v2);
    k_gemm1_wmma<<<(NN / 16) * (C1 / 64) / 8, 256, 0, stream>>>(x, W1, h1);
    k_nodelogits1<<<NN * H1n / 256, 256, 0, stream>>>(h1, as1, ad1, als1, ald1);
    k_gat1<<<NGr, 256, 0, stream>>>(h1, als1, ald1, ea, v1, b1, hagg1);
    k_bn_partial<C1><<<PB, C1, 0, stream>>>(hagg1, sum1, sq1);
    k_bn_final<C1><<<1, C1, 0, stream>>>(sum1, sq1, g1, be1, sc1, sh1);
    k_gemm2_wmma<<<(NN / 16) / 8, 256, 0, stream>>>(hagg1, W2, sc1, sh1, h2);
    k_nodelogits2<<<NN / 256, 256, 0, stream>>>(h2, as2, ad2, als2, ald2);
    k_gat2<<<NGr, HID, 0, stream>>>(h2, als2, ald2, ea, v2, b2, hagg2);
    k_bn_partial<HID><<<PB, HID, 0, stream>>>(hagg2, sum2, sq2);
    k_bn_final<HID><<<1, HID, 0, stream>>>(sum2, sq2, g2, be2, sc2, sh2);
    k_head<<<NGr, HID, 0, stream>>>(hagg2, sc2, sh2, gf, Wo1, bo1, Wo2, bo2,
                                    Wb1, bb1, Wb2, bb2, out);
}